// GCN_16638703304811
// MI455X (gfx1250) — compile-verified
//
#include <hip/hip_runtime.h>

typedef float v2f __attribute__((ext_vector_type(2)));
typedef float v8f __attribute__((ext_vector_type(8)));

#define GCN_N_NODES 50000
#define GCN_N_EDGES 800000
#define GCN_KDIM    512
#define GCN_LDA_S   516   // LDS row stride (floats): 512+4 pad -> conflict-free banks

// ---------------------------------------------------------------- utility ops
__global__ void gcn_zero_u32(unsigned* __restrict__ p, int n) {
    int i = blockIdx.x * blockDim.x + threadIdx.x;
    if (i < n) p[i] = 0u;
}

__global__ void gcn_count_deg(const int* __restrict__ dst, unsigned* __restrict__ deg,
                              int n_edges, int n_nodes) {
    int i = blockIdx.x * blockDim.x + threadIdx.x;
    int total = n_edges + n_nodes;
    if (i >= total) return;
    int d = (i < n_edges) ? dst[i] : (i - n_edges);   // self-loop tail
    atomicAdd(&deg[d], 1u);
}

__global__ void gcn_dinv(const unsigned* __restrict__ deg, float* __restrict__ dinv, int n) {
    int i = blockIdx.x * blockDim.x + threadIdx.x;
    if (i < n) {
        unsigned d = deg[i];
        dinv[i] = d ? rsqrtf((float)d) : 0.0f;
    }
}

// out[i] = bias[i & (C-1)]   (C is a power of two)
__global__ void gcn_bias_init(const float* __restrict__ b, float* __restrict__ out,
                              int total, int C) {
    int i = blockIdx.x * blockDim.x + threadIdx.x;
    if (i < total) out[i] = b[i & (C - 1)];
}

// ------------------------------------------------------------- WMMA fp32 GEMM
// C[M=grid*16, Ncols] = (relu_a ? relu(A) : A)[M, 512] @ B[512, Ncols]
// One block per 16-row M tile. A tile staged in LDS. Each wave owns 16x64 of C
// (4 f32 accumulators), so grid = M/16, blockDim = (Ncols/64)*32.
__global__ void gcn_gemm_wmma(const float* __restrict__ A, const float* __restrict__ B,
                              float* __restrict__ C, int Ncols, int relu_a) {
    __shared__ float As[16 * GCN_LDA_S];
    const int mbase = blockIdx.x * 16;

    // Stage 16x512 fp32 A tile into LDS (float4 per thread), fusing optional ReLU.
    for (int i = threadIdx.x; i < (16 * GCN_KDIM) / 4; i += blockDim.x) {
        int r = i >> 7;            // i / 128
        int c = (i & 127) << 2;    // (i % 128) * 4
        float4 v = *(const float4*)(A + (size_t)(mbase + r) * GCN_KDIM + c);
        if (relu_a) {
            v.x = fmaxf(v.x, 0.0f); v.y = fmaxf(v.y, 0.0f);
            v.z = fmaxf(v.z, 0.0f); v.w = fmaxf(v.w, 0.0f);
        }
        *(float4*)(&As[r * GCN_LDA_S + c]) = v;
    }
    __syncthreads();

    const int lane = threadIdx.x & 31;
    const int ln   = lane & 15;          // row (A) / col-in-tile (B,C)
    const int kh   = (lane >> 4) << 1;   // 0 for lanes 0-15, 2 for lanes 16-31
    const int nb   = (threadIdx.x >> 5) * 64;

    v8f acc0 = {}, acc1 = {}, acc2 = {}, acc3 = {};
    const float* Asp = &As[ln * GCN_LDA_S + kh];

    for (int k = 0; k < GCN_KDIM; k += 4) {
        // A fragment (16x4 f32): lane row = ln, holds K = k+kh, k+kh+1
        v2f a = *(const v2f*)(Asp + k);
        // B fragments (4x16 f32): VGPR0 = row k+kh, VGPR1 = row k+kh+1, col = nb+ln
        const float* brow = B + (size_t)(k + kh) * Ncols + nb + ln;
        v2f b0; b0.x = brow[0];  b0.y = brow[Ncols];
        v2f b1; b1.x = brow[16]; b1.y = brow[16 + Ncols];
        v2f b2; b2.x = brow[32]; b2.y = brow[32 + Ncols];
        v2f b3; b3.x = brow[48]; b3.y = brow[48 + Ncols];
        acc0 = __builtin_amdgcn_wmma_f32_16x16x4_f32(false, a, false, b0, (short)0, acc0, false, false);
        acc1 = __builtin_amdgcn_wmma_f32_16x16x4_f32(false, a, false, b1, (short)0, acc1, false, false);
        acc2 = __builtin_amdgcn_wmma_f32_16x16x4_f32(false, a, false, b2, (short)0, acc2, false, false);
        acc3 = __builtin_amdgcn_wmma_f32_16x16x4_f32(false, a, false, b3, (short)0, acc3, false, false);
    }

    // D layout: VGPR i -> row i (lanes 0-15) / row 8+i (lanes 16-31), col = ln
    const int ro = (lane >> 4) << 3;
    float* Cp = C + (size_t)(mbase + ro) * Ncols + nb + ln;
#pragma unroll
    for (int i = 0; i < 8; ++i) {
        Cp[(size_t)i * Ncols + 0]  = acc0[i];
        Cp[(size_t)i * Ncols + 16] = acc1[i];
        Cp[(size_t)i * Ncols + 32] = acc2[i];
        Cp[(size_t)i * Ncols + 48] = acc3[i];
    }
}

// --------------------------------------------------------- scatter aggregation
// out[dst] += feat[src] * dinv[src]*dinv[dst]  (out pre-initialized with bias)
// One thread per (edge, 4 channels). gshift = log2(C/4).
__global__ void gcn_aggregate(const float* __restrict__ feat,
                              const int* __restrict__ src, const int* __restrict__ dst,
                              const float* __restrict__ dinv,
                              float* __restrict__ out,
                              int n_edges, int n_nodes, int C, int gshift) {
    int tid = blockIdx.x * blockDim.x + threadIdx.x;
    int total = (n_edges + n_nodes) << gshift;
    if (tid >= total) return;
    int e = tid >> gshift;
    int g = (tid & ((1 << gshift) - 1)) << 2;
    int s, d;
    if (e < n_edges) { s = src[e]; d = dst[e]; }
    else             { s = e - n_edges; d = s; }   // self loop
    float sc = dinv[s] * dinv[d];
    float4 v = *(const float4*)(feat + (size_t)s * C + g);
    float* o = out + (size_t)d * C + g;
    atomicAdd(o + 0, v.x * sc);
    atomicAdd(o + 1, v.y * sc);
    atomicAdd(o + 2, v.z * sc);
    atomicAdd(o + 3, v.w * sc);
}

// ------------------------------------------------------------------- launcher
extern "C" void kernel_launch(void* const* d_in, const int* in_sizes, int n_in,
                              void* d_out, int out_size, void* d_ws, size_t ws_size,
                              hipStream_t stream) {
    const float* x  = (const float*)d_in[0];
    const int*   ei = (const int*)d_in[1];     // [2, 800000] int32 (JAX x64 off)
    const float* w1 = (const float*)d_in[2];
    const float* b1 = (const float*)d_in[3];
    const float* w2 = (const float*)d_in[4];
    const float* b2 = (const float*)d_in[5];
    float* out = (float*)d_out;

    char* ws = (char*)d_ws;
    unsigned* deg  = (unsigned*)(ws + 0);                 //   200,000 B
    float*    dinv = (float*)   (ws + 200192);            //   200,000 B
    float*    h    = (float*)   (ws + 400384);            // 102,400,000 B
    float*    agg1 = (float*)   (ws + 102800384ULL);      // 102,400,000 B
    float*    t2   = h;  // h is dead after aggregation 1; reuse for GEMM2 output

    const int* srcp = ei;
    const int* dstp = ei + GCN_N_EDGES;

    const int E = GCN_N_EDGES, N = GCN_N_NODES;

    // degree + dinv
    gcn_zero_u32<<<(N + 255) / 256, 256, 0, stream>>>(deg, N);
    gcn_count_deg<<<(E + N + 255) / 256, 256, 0, stream>>>(dstp, deg, E, N);
    gcn_dinv<<<(N + 255) / 256, 256, 0, stream>>>(deg, dinv, N);

    // layer 1: h = x @ w1 ; agg1 = b1 + scatter(h*norm) ; relu folded into GEMM2
    gcn_gemm_wmma<<<N / 16, 256, 0, stream>>>(x, w1, h, 512, 0);
    gcn_bias_init<<<(N * 512 + 255) / 256, 256, 0, stream>>>(b1, agg1, N * 512, 512);
    {
        long long tot = (long long)(E + N) << 7;  // *128 groups (512/4)
        gcn_aggregate<<<(int)((tot + 255) / 256), 256, 0, stream>>>(
            h, srcp, dstp, dinv, agg1, E, N, 512, 7);
    }

    // layer 2: t2 = relu(agg1) @ w2 ; out = b2 + scatter(t2*norm)
    gcn_bias_init<<<(N * 256 + 255) / 256, 256, 0, stream>>>(b2, out, N * 256, 256);
    gcn_gemm_wmma<<<N / 16, 128, 0, stream>>>(agg1, w2, t2, 256, 1);
    {
        long long tot = (long long)(E + N) << 6;  // *64 groups (256/4)
        gcn_aggregate<<<(int)((tot + 255) / 256), 256, 0, stream>>>(
            t2, srcp, dstp, dinv, out, E, N, 256, 6);
    }
}